// GraphConvEncoder_62483184222413
// MI455X (gfx1250) — compile-verified
//
#include <hip/hip_runtime.h>
#include <math.h>

// Problem constants (from reference)
#define NNODES 50000
#define NEDGES 800000
#define DIM    128
#define KSEL   40000
#define NF     (NNODES*DIM)
#define NSLABS (NNODES/16)   // 3125, exact
#define LDW    132           // padded LDS row stride for W (bank-conflict avoidance)

typedef __attribute__((ext_vector_type(2))) float v2f;
typedef __attribute__((ext_vector_type(8))) float v8f;

// ---- order-preserving float <-> u32 mapping (for top-k / max via integer atomics)
__device__ __forceinline__ unsigned f2ord(float f) {
  unsigned b = __float_as_uint(f);
  return (b & 0x80000000u) ? ~b : (b | 0x80000000u);
}
__device__ __forceinline__ float ord2f(unsigned u) {
  unsigned b = (u & 0x80000000u) ? (u & 0x7FFFFFFFu) : ~u;
  return __uint_as_float(b);
}

// misc layout (32-bit words): [0]=thresh_u [1]=quota [2]=tie_counter [3]=gmax_u
//                             [4]=pnorm(f) [5]=S(f)  [8..135]=V[128](f)

// ---------------- K0: init workspace ----------------
__global__ void k_init(float* __restrict__ agg, float* __restrict__ deg,
                       float* __restrict__ misc) {
  long long stride = (long long)gridDim.x * blockDim.x;
  long long total  = (long long)NF + NNODES + 256;
  for (long long i = (long long)blockIdx.x * blockDim.x + threadIdx.x;
       i < total; i += stride) {
    if (i < NF)                agg[i] = 0.0f;
    else if (i < NF + NNODES)  deg[i - NF] = 1.0f;   // self-loop degree
    else                       misc[i - NF - NNODES] = 0.0f;
  }
}

// ---------------- K1: xw = x @ W  (fp32 WMMA 16x16x4) ----------------
__global__ __launch_bounds__(256) void k_gemm(const float* __restrict__ x,
                                              const float* __restrict__ W,
                                              float* __restrict__ xw) {
  __shared__ float Wl[DIM * LDW];   // 128x128 W, padded rows
  for (int i4 = threadIdx.x; i4 < (DIM * DIM) / 4; i4 += blockDim.x) {
    int idx = i4 * 4;
    int r = idx >> 7, c = idx & 127;
    float4 v = *(const float4*)(W + idx);
    *(float4*)(&Wl[r * LDW + c]) = v;
  }
  __syncthreads();

  int wave = threadIdx.x >> 5;
  int lane = threadIdx.x & 31;
  int slab = blockIdx.x * 8 + wave;
  if (slab >= NSLABS) return;           // wave-uniform exit; EXEC stays all-1s

  int r0   = slab * 16;
  int m    = lane & 15;                 // row (A) / col (B,D) within tile
  int half = lane >> 4;                 // lane group selects K pair / M half

  const float* xrow = x + (long long)(r0 + m) * DIM;
  v8f acc[8] = {};

  for (int k4 = 0; k4 < DIM / 4; ++k4) {
    int kk = k4 * 4 + half * 2;         // A: VGPR v holds K = v + 2*group
    v2f a;
    a.x = xrow[kk];
    a.y = xrow[kk + 1];
#pragma unroll
    for (int t = 0; t < 8; ++t) {       // 8 N-tiles -> 16x128 slab
      v2f b;                            // B: VGPR v holds K = v + 2*group
      b.x = Wl[kk * LDW + t * 16 + m];
      b.y = Wl[(kk + 1) * LDW + t * 16 + m];
      acc[t] = __builtin_amdgcn_wmma_f32_16x16x4_f32(
          false, a, false, b, (short)0, acc[t], false, false);
    }
  }
#pragma unroll
  for (int t = 0; t < 8; ++t) {
#pragma unroll
    for (int v = 0; v < 8; ++v) {       // D: VGPR v, lane group g -> M = v + 8g
      xw[(long long)(r0 + v + 8 * half) * DIM + t * 16 + m] = acc[t][v];
    }
  }
}

// ---------------- K2: degree (deg pre-init to 1.0 for self loop) ----------------
__global__ void k_deg(const long long* __restrict__ ei, float* __restrict__ deg) {
  int e = blockIdx.x * blockDim.x + threadIdx.x;
  if (e < NEDGES) {
    int dst = (int)ei[NEDGES + e];
    atomicAdd(&deg[dst], 1.0f);
  }
}

// ---------------- K3: edge scatter: agg[dst] += xw[src]*norm ----------------
__global__ __launch_bounds__(256) void k_scatter(const long long* __restrict__ ei,
                                                 const float* __restrict__ xw,
                                                 const float* __restrict__ deg,
                                                 float* __restrict__ agg) {
  int wave = threadIdx.x >> 5, lane = threadIdx.x & 31;
  long long e = (long long)blockIdx.x * 8 + wave;
  if (e >= NEDGES) return;
  int src = (int)ei[e];
  int dst = (int)ei[NEDGES + e];
  float nrm = rsqrtf(deg[src]) * rsqrtf(deg[dst]);
  float4 v = *(const float4*)(xw + (long long)src * DIM + lane * 4);
  float* ap = agg + (long long)dst * DIM + lane * 4;
  atomicAdd(ap + 0, v.x * nrm);
  atomicAdd(ap + 1, v.y * nrm);
  atomicAdd(ap + 2, v.z * nrm);
  atomicAdd(ap + 3, v.w * nrm);
}

// ---------------- K4: h = relu(agg + xw/deg + b) (in-place into agg) ----------------
__global__ void k_h(const float* __restrict__ xw, const float* __restrict__ deg,
                    const float* __restrict__ b, float* __restrict__ h) {
  long long i = (long long)blockIdx.x * blockDim.x + threadIdx.x;
  if (i >= (long long)NF / 4) return;
  long long idx = i * 4;
  int n = (int)(idx >> 7);
  int c = (int)(idx & 127);
  float inv = 1.0f / deg[n];            // dis*dis
  float4 xv = *(const float4*)(xw + idx);
  float4 av = *(const float4*)(h + idx);
  float4 bv = *(const float4*)(b + c);
  float4 r;
  r.x = fmaxf(av.x + xv.x * inv + bv.x, 0.0f);
  r.y = fmaxf(av.y + xv.y * inv + bv.y, 0.0f);
  r.z = fmaxf(av.z + xv.z * inv + bv.z, 0.0f);
  r.w = fmaxf(av.w + xv.w * inv + bv.w, 0.0f);
  *(float4*)(h + idx) = r;
}

// ---------------- K5: ||p|| ----------------
__global__ void k_pnorm(const float* __restrict__ p, float* __restrict__ miscF) {
  __shared__ float red[128];
  float v = p[threadIdx.x];
  red[threadIdx.x] = v * v;
  __syncthreads();
  for (int s = 64; s; s >>= 1) {
    if ((int)threadIdx.x < s) red[threadIdx.x] += red[threadIdx.x + s];
    __syncthreads();
  }
  if (threadIdx.x == 0) miscF[4] = sqrtf(red[0]);
}

// ---------------- K6: per-node score (tanh) + gate dot, fused ----------------
__global__ __launch_bounds__(256) void k_score(const float* __restrict__ h,
                                               const float* __restrict__ p,
                                               const float* __restrict__ gw,
                                               const float* __restrict__ gb,
                                               const float* __restrict__ miscF,
                                               float* __restrict__ score,
                                               float* __restrict__ gate,
                                               unsigned* __restrict__ su) {
  int wave = threadIdx.x >> 5, lane = threadIdx.x & 31;
  int n = blockIdx.x * 8 + wave;
  if (n >= NNODES) return;
  float4 hv = *(const float4*)(h + (long long)n * DIM + lane * 4);
  float4 pv = *(const float4*)(p + lane * 4);
  float4 gv = *(const float4*)(gw + lane * 4);
  float dp = hv.x * pv.x + hv.y * pv.y + hv.z * pv.z + hv.w * pv.w;
  float dg = hv.x * gv.x + hv.y * gv.y + hv.z * gv.z + hv.w * gv.w;
  for (int off = 16; off; off >>= 1) {
    dp += __shfl_down(dp, off);
    dg += __shfl_down(dg, off);
  }
  if (lane == 0) {
    float s = tanhf(dp / miscF[4]);
    score[n] = s;
    gate[n]  = s * dg + gb[0];          // gate of hp = s*h
    su[n]    = f2ord(s);
  }
}

// ---------------- K7: exact K-th largest via 4-pass radix (single block) ----------------
__global__ __launch_bounds__(1024) void k_radix(const unsigned* __restrict__ su,
                                                unsigned* __restrict__ miscU) {
  __shared__ unsigned hist[256];
  __shared__ unsigned sPrefix, sRemaining;
  if (threadIdx.x == 0) { sPrefix = 0u; sRemaining = (unsigned)KSEL; }
  __syncthreads();
  for (int pass = 0; pass < 4; ++pass) {
    int shift = 24 - 8 * pass;
    if (threadIdx.x < 256) hist[threadIdx.x] = 0u;
    __syncthreads();
    unsigned prefix = sPrefix;
    unsigned himask = (pass == 0) ? 0u : (0xFFFFFFFFu << (shift + 8));
    for (int i = threadIdx.x; i < NNODES; i += blockDim.x) {
      unsigned u = su[i];
      if ((u & himask) == (prefix & himask))
        atomicAdd(&hist[(u >> shift) & 0xFF], 1u);
    }
    __syncthreads();
    if (threadIdx.x == 0) {
      unsigned rem = sRemaining, cum = 0;
      int bsel = 0;
      for (int bb = 255; bb >= 0; --bb) {
        unsigned c = hist[bb];
        if (cum + c >= rem) { bsel = bb; break; }
        cum += c;
      }
      sPrefix    = prefix | ((unsigned)bsel << shift);
      sRemaining = rem - cum;
    }
    __syncthreads();
  }
  if (threadIdx.x == 0) {
    miscU[0] = sPrefix;      // threshold key (K-th largest score, order-mapped)
    miscU[1] = sRemaining;   // how many ties at threshold to accept
  }
}

// ---------------- K8: materialize selection mask ----------------
__global__ void k_select(const unsigned* __restrict__ su, unsigned* __restrict__ sel,
                         unsigned* __restrict__ miscU) {
  int n = blockIdx.x * blockDim.x + threadIdx.x;
  if (n >= NNODES) return;
  unsigned u = su[n], th = miscU[0], q = miscU[1];
  unsigned s;
  if (u > th)       s = 1u;
  else if (u == th) s = (atomicAdd(&miscU[2], 1u) < q) ? 1u : 0u;
  else              s = 0u;
  sel[n] = s;
}

// ---------------- K9: max gate over selected (monotonic-u32 atomicMax) ----------------
__global__ void k_gmax(const float* __restrict__ gate, const unsigned* __restrict__ sel,
                       unsigned* __restrict__ miscU) {
  int n = blockIdx.x * blockDim.x + threadIdx.x;
  if (n >= NNODES) return;
  if (sel[n]) atomicMax(&miscU[3], f2ord(gate[n]));
}

// ---------------- K10: V = sum exp(g-gmax) * s*h ;  S = sum exp(g-gmax) ----------------
__global__ __launch_bounds__(256) void k_attn(const float* __restrict__ h,
                                              const float* __restrict__ score,
                                              const float* __restrict__ gate,
                                              const unsigned* __restrict__ sel,
                                              const unsigned* __restrict__ miscU,
                                              float* __restrict__ miscF) {
  int wave = threadIdx.x >> 5, lane = threadIdx.x & 31;
  float gmax = ord2f(miscU[3]);
  float accs = 0.0f;
  float4 accv = make_float4(0.f, 0.f, 0.f, 0.f);
  int base = (blockIdx.x * 8 + wave) * 32;   // 32 nodes per wave
  for (int j = 0; j < 32; ++j) {
    int n = base + j;
    if (n >= NNODES) break;                  // wave-uniform
    if (!sel[n]) continue;                   // wave-uniform
    float w  = expf(gate[n] - gmax);
    float ws = w * score[n];                 // hp = score * h
    float4 hv = *(const float4*)(h + (long long)n * DIM + lane * 4);
    accs   += w;
    accv.x += ws * hv.x;
    accv.y += ws * hv.y;
    accv.z += ws * hv.z;
    accv.w += ws * hv.w;
  }
  float* V = miscF + 8;
  atomicAdd(&V[lane * 4 + 0], accv.x);
  atomicAdd(&V[lane * 4 + 1], accv.y);
  atomicAdd(&V[lane * 4 + 2], accv.z);
  atomicAdd(&V[lane * 4 + 3], accv.w);
  if (lane == 0) atomicAdd(&miscF[5], accs);
}

// ---------------- K11: out = V / S ----------------
__global__ void k_final(const float* __restrict__ miscF, float* __restrict__ out) {
  int c = threadIdx.x;
  if (c < DIM) out[c] = miscF[8 + c] / miscF[5];
}

extern "C" void kernel_launch(void* const* d_in, const int* in_sizes, int n_in,
                              void* d_out, int out_size, void* d_ws, size_t ws_size,
                              hipStream_t stream) {
  (void)in_sizes; (void)n_in; (void)out_size; (void)ws_size;
  const float*     x  = (const float*)d_in[0];
  const long long* ei = (const long long*)d_in[1];   // int64 edge_index [2,E]
  const float*     W  = (const float*)d_in[2];
  const float*     b  = (const float*)d_in[3];
  const float*     p  = (const float*)d_in[4];
  const float*     gw = (const float*)d_in[5];
  const float*     gb = (const float*)d_in[6];
  float*           out = (float*)d_out;

  // workspace carve-up (~52.2 MB)
  float*    xw    = (float*)d_ws;          // N*128
  float*    h     = xw + NF;               // N*128 (agg, then h in-place)
  float*    deg   = h + NF;                // N
  float*    score = deg + NNODES;          // N
  float*    gate  = score + NNODES;        // N
  unsigned* su    = (unsigned*)(gate + NNODES);  // N
  unsigned* sel   = su + NNODES;                 // N
  float*    miscF = (float*)(sel + NNODES);      // 256 words
  unsigned* miscU = (unsigned*)miscF;

  k_init   <<<4096, 256, 0, stream>>>(h, deg, miscF);
  k_gemm   <<<(NSLABS + 7) / 8, 256, 0, stream>>>(x, W, xw);
  k_deg    <<<(NEDGES + 255) / 256, 256, 0, stream>>>(ei, deg);
  k_scatter<<<(NEDGES + 7) / 8, 256, 0, stream>>>(ei, xw, deg, h);
  k_h      <<<NF / 4 / 256, 256, 0, stream>>>(xw, deg, b, h);
  k_pnorm  <<<1, 128, 0, stream>>>(p, miscF);
  k_score  <<<(NNODES + 7) / 8, 256, 0, stream>>>(h, p, gw, gb, miscF, score, gate, su);
  k_radix  <<<1, 1024, 0, stream>>>(su, miscU);
  k_select <<<(NNODES + 255) / 256, 256, 0, stream>>>(su, sel, miscU);
  k_gmax   <<<(NNODES + 255) / 256, 256, 0, stream>>>(gate, sel, miscU);
  k_attn   <<<(NNODES + 255) / 256, 256, 0, stream>>>(h, score, gate, sel, miscU, miscF);
  k_final  <<<1, 128, 0, stream>>>(miscF, out);
}